// GraphConv_66125316489414
// MI455X (gfx1250) — compile-verified
//
#include <hip/hip_runtime.h>

#define N_NODES 50000
#define N_EDGES 800000
#define IC 2
#define OC 2
#define DIN 128
#define DOUT 128
#define KHOPS 3

typedef float v2f __attribute__((ext_vector_type(2)));
typedef float v8f __attribute__((ext_vector_type(8)));

// ---------------- degree / normalization ----------------
__global__ void degree_kernel(const int* __restrict__ row, float* __restrict__ deg) {
    int e = blockIdx.x * blockDim.x + threadIdx.x;
    if (e < N_EDGES) atomicAdd(&deg[row[e]], 1.0f);
}

__global__ void dinv_kernel(float* __restrict__ deg) {
    int n = blockIdx.x * blockDim.x + threadIdx.x;
    if (n < N_NODES) {
        float d = deg[n];
        deg[n] = d > 0.0f ? rsqrtf(d) : 0.0f;
    }
}

__global__ void val_kernel(const int* __restrict__ row, const int* __restrict__ col,
                           const float* __restrict__ dinv, float* __restrict__ val) {
    int e = blockIdx.x * blockDim.x + threadIdx.x;
    if (e < N_EDGES) val[e] = dinv[row[e]] * dinv[col[e]];
}

// ---------------- scale W by params, pre-swizzled for WMMA B loads ----------------
// Wm layout per (m,i,j): element (K=d, N=f) stored at ((d>>1)*DOUT + f)*2 + (d&1)
// so a lane's (K=k+2*hi, K=k+2*hi+1) pair for column f is one contiguous float2.
__global__ void wscale_kernel(const float* __restrict__ W, const float* __restrict__ params,
                              float* __restrict__ Wm) {
    int t = blockIdx.x * blockDim.x + threadIdx.x;  // < (K+1)*OC*IC*DIN*DOUT = 262144
    int f = t & 127;
    int d = (t >> 7) & 127;
    int j = (t >> 14) & 1;
    int i = (t >> 15) & 1;
    int m = t >> 16;
    float p = params[(i * IC + j) * (KHOPS + 1) + m];
    float w = W[(((size_t)(i * IC + j)) << 14) + (d << 7) + f];
    size_t base = ((size_t)((m * OC + i) * IC + j)) << 14;
    Wm[base + (((d >> 1) * DOUT + f) << 1) + (d & 1)] = p * w;
}

// ---------------- SpMM scatter: Hout[j, row[e], :] += val[e] * Hin[j, col[e], :] ----------------
// one wave per (edge, channel); 32 lanes x float4 = 128 columns
__global__ void spmm_kernel(const int* __restrict__ row, const int* __restrict__ col,
                            const float* __restrict__ val,
                            const float* __restrict__ Hin, float* __restrict__ Hout) {
    int wave = (blockIdx.x * blockDim.x + threadIdx.x) >> 5;
    int lane = threadIdx.x & 31;
    if (wave >= N_EDGES * IC) return;
    int e = wave % N_EDGES;
    int j = wave / N_EDGES;
    int r = row[e];
    int c = col[e];
    float v = val[e];
    const float4* src = (const float4*)(Hin + ((size_t)j * N_NODES + c) * DIN);
    float*       dst = Hout + ((size_t)j * N_NODES + r) * DIN;
    float4 g = src[lane];
    atomicAdd(dst + lane * 4 + 0, v * g.x);
    atomicAdd(dst + lane * 4 + 1, v * g.y);
    atomicAdd(dst + lane * 4 + 2, v * g.z);
    atomicAdd(dst + lane * 4 + 3, v * g.w);
}

// ---------------- WMMA GEMM: out[i, strip, :] (+)= sum_j H[j, strip, :] @ Wm[m,i,j] ----------------
// one wave per (16-row strip, output channel i); 8 x (16x16) f32 accum tiles = 128 cols
__global__ void gemm_kernel(const float* __restrict__ H, const float* __restrict__ Wm,
                            float* __restrict__ out, int accum) {
    const int n_tasks = (N_NODES / 16) * OC;  // 6250
    int lane = threadIdx.x & 31;
    int wave = blockIdx.x * (blockDim.x >> 5) + (threadIdx.x >> 5);
    if (wave >= n_tasks) return;
    int i = wave & 1;
    int strip = wave >> 1;
    int row_base = strip * 16;
    int hi = lane >> 4;   // lane half selects K offset (A) / K offset (B) / M offset (C)
    int lo = lane & 15;   // M index for A, N index for B/C

    v8f acc[8];
    v8f zero = {};
#pragma unroll
    for (int t = 0; t < 8; ++t) acc[t] = zero;

    for (int j = 0; j < IC; ++j) {
        const float* A = H + ((size_t)j * N_NODES + row_base) * DIN;
        const v2f*   B = (const v2f*)(Wm + (((size_t)i * IC + j) << 14));
        for (int k = 0; k < DIN; k += 4) {
            // A 16x4 f32 tile: lane lo = M, VGPR pair = K = {k+2*hi, k+2*hi+1}
            v2f a = *(const v2f*)(A + lo * DIN + k + 2 * hi);
            int kp = (k >> 1) + hi;  // K-pair row in swizzled B
#pragma unroll
            for (int nt = 0; nt < 8; ++nt) {
                // B 4x16 f32 tile: lane lo = N, VGPR pair = K = {k+2*hi, k+2*hi+1}
                v2f b = B[kp * DOUT + nt * 16 + lo];
                acc[nt] = __builtin_amdgcn_wmma_f32_16x16x4_f32(
                    false, a, false, b, (short)0, acc[nt], false, false);
            }
        }
    }

    // C/D 16x16 f32: VGPR r, lane -> (M = r + 8*hi, N = lo), tile nt -> cols nt*16..
    float* O = out + ((size_t)i * N_NODES + row_base) * DOUT;
#pragma unroll
    for (int nt = 0; nt < 8; ++nt) {
#pragma unroll
        for (int r = 0; r < 8; ++r) {
            size_t idx = (size_t)(r + 8 * hi) * DOUT + nt * 16 + lo;
            float prev = accum ? O[idx] : 0.0f;
            O[idx] = prev + acc[nt][r];
        }
    }
}

extern "C" void kernel_launch(void* const* d_in, const int* in_sizes, int n_in,
                              void* d_out, int out_size, void* d_ws, size_t ws_size,
                              hipStream_t stream) {
    const float* x          = (const float*)d_in[0];  // [IC, N, DIN]
    const int*   edge_index = (const int*)  d_in[1];  // [2, E]
    const float* W          = (const float*)d_in[2];  // [OC, IC, DIN, DOUT]
    const float* params     = (const float*)d_in[3];  // [OC, IC, K+1]
    float* out = (float*)d_out;                       // [OC, N, DOUT]

    const int* row = edge_index;            // edge_index[0]
    const int* col = edge_index + N_EDGES;  // edge_index[1]

    // workspace layout (floats)
    float* ws = (float*)d_ws;
    size_t off = 0;
    float* deg = ws + off; off += 50176;                               // deg -> dinv (padded)
    float* val = ws + off; off += N_EDGES;                             // edge weights
    float* Wm  = ws + off; off += (size_t)(KHOPS + 1) * OC * IC * DIN * DOUT;
    float* H0  = ws + off; off += (size_t)IC * N_NODES * DIN;
    float* H1  = ws + off;

    const size_t hbytes = (size_t)IC * N_NODES * DIN * sizeof(float);

    hipMemsetAsync(deg, 0, N_NODES * sizeof(float), stream);
    degree_kernel<<<(N_EDGES + 255) / 256, 256, 0, stream>>>(row, deg);
    dinv_kernel<<<(N_NODES + 255) / 256, 256, 0, stream>>>(deg);
    val_kernel<<<(N_EDGES + 255) / 256, 256, 0, stream>>>(row, col, deg, val);
    wscale_kernel<<<((KHOPS + 1) * OC * IC * DIN * DOUT) / 256, 256, 0, stream>>>(W, params, Wm);
    hipMemcpyAsync(H0, x, hbytes, hipMemcpyDeviceToDevice, stream);

    const int gemm_blocks = ((N_NODES / 16) * OC + 7) / 8;        // 8 waves/block
    const int spmm_blocks = (int)(((long long)N_EDGES * IC + 7) / 8);

    float* Hc = H0;
    float* Hn = H1;
    for (int m = 0; m <= KHOPS; ++m) {
        gemm_kernel<<<gemm_blocks, 256, 0, stream>>>(
            Hc, Wm + ((size_t)m * OC * IC * DIN * DOUT), out, m > 0 ? 1 : 0);
        if (m < KHOPS) {
            hipMemsetAsync(Hn, 0, hbytes, stream);
            spmm_kernel<<<spmm_blocks, 256, 0, stream>>>(row, col, val, Hc, Hn);
            float* t = Hc; Hc = Hn; Hn = t;
        }
    }
}